// MultiscaleMessagePassing_90374701842961
// MI455X (gfx1250) — compile-verified
//
#include <hip/hip_runtime.h>
#include <hip/hip_bf16.h>

// Problem constants (match reference)
#define HD        256
#define N_FINE    262144
#define E_FINE    524288
#define N_COARSE  65536
#define E_COARSE  262144

#define TM        32            // fine-node rows per workgroup tile
#define NSUB      (TM/16)       // 16-row WMMA sub-tiles per wave
#define LDSS      260           // padded LDS row stride (floats): 260%64==4 -> conflict-free A reads

#define NODE_SUM  ((size_t)N_COARSE * HD)          // 16,777,216 floats
#define EDGE_SUM  ((size_t)E_COARSE * HD)          // 67,108,864 floats
#define SUM_TOT   (NODE_SUM + EDGE_SUM)            // 83,886,080 floats

typedef __attribute__((ext_vector_type(2))) float v2f;
typedef __attribute__((ext_vector_type(8))) float v8f;

__device__ __forceinline__ float elu1(float v) {
    return v > 0.0f ? v : (expf(v) - 1.0f);
}

// K=256 GEMM panel: D(16x16 per subtile) += A(16x256, LDS) * B(256x16, global W slice)
// A        : pre-offset LDS pointer  (buf + l16*LDSS + 2*g)
// Wp       : pre-offset weight ptr   (W + 2*g*HD + col)
// acc[s][p]: 2 parity accumulators per subtile -> 4 independent WMMA chains
__device__ __forceinline__ void gemm_panel(const float* __restrict__ A,
                                           const float* __restrict__ Wp,
                                           v8f acc[NSUB][2]) {
#pragma unroll 4
    for (int k = 0; k < HD; k += 4) {
        v2f b;
        b.x = Wp[k * HD];            // K = k + 2g + 0
        b.y = Wp[(k + 1) * HD];      // K = k + 2g + 1
        const int p = (k >> 2) & 1;
#pragma unroll
        for (int s = 0; s < NSUB; ++s) {
            v2f a;
            const float* ap = A + s * 16 * LDSS + k;
            a.x = ap[0];
            a.y = ap[1];
            acc[s][p] = __builtin_amdgcn_wmma_f32_16x16x4_f32(
                false, a, false, b, (short)0, acc[s][p], false, false);
        }
    }
}

// ---------------------------------------------------------------------------
// Fused node MLP (Linear-ELU-Linear-ELU-Linear) + residual + atomic scatter-sum
// ---------------------------------------------------------------------------
__global__ void __launch_bounds__(512)
node_mlp_scatter(const float* __restrict__ x, const float* __restrict__ dist,
                 const int* __restrict__ f2c,
                 const float* __restrict__ W1, const float* __restrict__ b1,
                 const float* __restrict__ W2, const float* __restrict__ b2,
                 const float* __restrict__ W3, const float* __restrict__ b3,
                 float* __restrict__ nsum, float* __restrict__ ncnt) {
    extern __shared__ float buf[];        // TM x LDSS, ping-pongs x -> h1 -> h2
    __shared__ float sdist[TM];
    __shared__ int   sidx[TM];

    const int tid  = threadIdx.x;
    const int wave = tid >> 5;
    const int lane = tid & 31;
    const int g    = lane >> 4;           // half-wave group
    const int l16  = lane & 15;
    const int col  = wave * 16 + l16;     // this lane's output column
    const size_t row0 = (size_t)blockIdx.x * TM;

    // ---- stage x tile (TM x 256) into LDS, plus dist / fine2coarse index
    for (int i = tid; i < TM * (HD / 4); i += 512) {
        const int r  = i >> 6;            // HD/4 == 64 float4 per row
        const int c4 = i & 63;
        float4 v = ((const float4*)(x + (row0 + r) * HD))[c4];
        *(float4*)(buf + r * LDSS + c4 * 4) = v;
    }
    if (tid < TM) { sdist[tid] = dist[row0 + tid]; sidx[tid] = f2c[row0 + tid]; }
    __syncthreads();

    const float* Aptr = buf + l16 * LDSS + 2 * g;
    const int woff = 2 * g * HD + col;
    v8f acc[NSUB][2];
    const v8f vz = {};

    // ================= layer 1: h1 = elu(cat(x,d) @ W1 + b1) =================
#pragma unroll
    for (int s = 0; s < NSUB; ++s) { acc[s][0] = vz; acc[s][1] = vz; }
    gemm_panel(Aptr, W1 + woff, acc);
    __syncthreads();                       // all waves done reading x
    {
        const float wl = W1[(size_t)HD * HD + col];   // distance row of W1
        const float bb = b1[col];
#pragma unroll
        for (int s = 0; s < NSUB; ++s)
#pragma unroll
            for (int v = 0; v < 8; ++v) {
                const int r = s * 16 + v + 8 * g;
                float val = acc[s][0][v] + acc[s][1][v] + sdist[r] * wl + bb;
                buf[r * LDSS + col] = elu1(val);
            }
    }
    __syncthreads();

    // ================= layer 2: h2 = elu(h1 @ W2 + b2) =======================
#pragma unroll
    for (int s = 0; s < NSUB; ++s) { acc[s][0] = vz; acc[s][1] = vz; }
    gemm_panel(Aptr, W2 + woff, acc);
    __syncthreads();                       // all waves done reading h1
    {
        const float bb = b2[col];
#pragma unroll
        for (int s = 0; s < NSUB; ++s)
#pragma unroll
            for (int v = 0; v < 8; ++v) {
                const int r = s * 16 + v + 8 * g;
                buf[r * LDSS + col] = elu1(acc[s][0][v] + acc[s][1][v] + bb);
            }
    }
    __syncthreads();

    // ======== layer 3 + residual + scatter-sum onto coarse nodes =============
#pragma unroll
    for (int s = 0; s < NSUB; ++s) { acc[s][0] = vz; acc[s][1] = vz; }
    gemm_panel(Aptr, W3 + woff, acc);
    {
        const float bb = b3[col];
#pragma unroll
        for (int s = 0; s < NSUB; ++s)
#pragma unroll
            for (int v = 0; v < 8; ++v) {
                const int r = s * 16 + v + 8 * g;
                float val = acc[s][0][v] + acc[s][1][v] + bb
                          + x[(row0 + r) * HD + col];          // residual
                atomicAdd(&nsum[(size_t)sidx[r] * HD + col], val);
            }
    }
    if (tid < TM) atomicAdd(&ncnt[sidx[tid]], 1.0f);
}

// ---------------------------------------------------------------------------
// Edge feature scatter-sum: 64 threads per fine edge, float4 loads
// ---------------------------------------------------------------------------
__global__ void __launch_bounds__(256)
edge_scatter(const float* __restrict__ ea, const int* __restrict__ f2ce,
             float* __restrict__ esum, float* __restrict__ ecnt) {
    const size_t e = (size_t)blockIdx.x * 4 + (threadIdx.x >> 6);
    const int q = threadIdx.x & 63;
    const int c = f2ce[e];
    float4 v = ((const float4*)(ea + e * HD))[q];
    float* dst = esum + (size_t)c * HD + q * 4;
    atomicAdd(dst + 0, v.x);
    atomicAdd(dst + 1, v.y);
    atomicAdd(dst + 2, v.z);
    atomicAdd(dst + 3, v.w);
    if (q == 0) atomicAdd(&ecnt[c], 1.0f);
}

// ---------------------------------------------------------------------------
// Zero accumulators/counters, bit-copy coarse_edge_index into output tail
// ---------------------------------------------------------------------------
__global__ void __launch_bounds__(256)
init_kernel(float* __restrict__ out, float* __restrict__ cnts,
            const int* __restrict__ cei) {
    const size_t i = (size_t)blockIdx.x * blockDim.x + threadIdx.x;
    if (i < SUM_TOT / 4) ((float4*)out)[i] = make_float4(0.f, 0.f, 0.f, 0.f);
    if (i < (size_t)(N_COARSE + E_COARSE)) cnts[i] = 0.f;
    if (i < (size_t)2 * E_COARSE) ((int*)(out + SUM_TOT))[i] = cei[i];
}

// ---------------------------------------------------------------------------
// sums -> means: divide each coarse row by max(count, 1)  (true fp32 division)
// ---------------------------------------------------------------------------
__global__ void __launch_bounds__(256)
finalize_kernel(float* __restrict__ out, const float* __restrict__ ncnt,
                const float* __restrict__ ecnt) {
    const size_t i4 = (size_t)blockIdx.x * blockDim.x + threadIdx.x;
    if (i4 >= SUM_TOT / 4) return;
    const size_t i = i4 * 4;
    float c;
    if (i < NODE_SUM) c = fmaxf(ncnt[i >> 8], 1.0f);
    else              c = fmaxf(ecnt[(i - NODE_SUM) >> 8], 1.0f);
    float4 v = ((float4*)out)[i4];
    v.x /= c; v.y /= c; v.z /= c; v.w /= c;
    ((float4*)out)[i4] = v;
}

extern "C" void kernel_launch(void* const* d_in, const int* in_sizes, int n_in,
                              void* d_out, int out_size, void* d_ws, size_t ws_size,
                              hipStream_t stream) {
    const float* x    = (const float*)d_in[0];
    const float* ea   = (const float*)d_in[1];
    const float* dist = (const float*)d_in[2];
    const int*   f2c  = (const int*)d_in[3];
    const int*   f2ce = (const int*)d_in[4];
    const int*   cei  = (const int*)d_in[5];
    const float* W1   = (const float*)d_in[6];
    const float* b1   = (const float*)d_in[7];
    const float* W2   = (const float*)d_in[8];
    const float* b2   = (const float*)d_in[9];
    const float* W3   = (const float*)d_in[10];
    const float* b3   = (const float*)d_in[11];

    float* out  = (float*)d_out;
    float* nsum = out;                  // [N_COARSE * H]
    float* esum = out + NODE_SUM;       // [E_COARSE * H]
    float* cnts = (float*)d_ws;         // node counts then edge counts
    float* ncnt = cnts;
    float* ecnt = cnts + N_COARSE;

    {   // init: zero sums + counts, copy edge index
        const size_t n = SUM_TOT / 4;   // 20,971,520 float4 groups (covers all init work)
        init_kernel<<<(unsigned)((n + 255) / 256), 256, 0, stream>>>(out, cnts, cei);
    }

    edge_scatter<<<E_FINE / 4, 256, 0, stream>>>(ea, f2ce, esum, ecnt);

    node_mlp_scatter<<<N_FINE / TM, 512, TM * LDSS * sizeof(float), stream>>>(
        x, dist, f2c, W1, b1, W2, b2, W3, b3, nsum, ncnt);

    finalize_kernel<<<(unsigned)((SUM_TOT / 4 + 255) / 256), 256, 0, stream>>>(
        out, ncnt, ecnt);
}